// PointConv_80272938762654
// MI455X (gfx1250) — compile-verified
//
#include <hip/hip_runtime.h>
#include <math.h>

// ---------------------------------------------------------------------------
// PointConv-style pipeline for MI455X (gfx1250, wave32).
//
// Roofline: total working set ~2.2MB (L2-resident), FLOPs ~0.4 GFLOP.
// Nothing here stresses 23.3 TB/s HBM; the cost is (a) the inherently
// sequential 2048-step FPS loop and (b) three 4096x64x32 f32 GEMMs, which we
// run on the WMMA pipe with V_WMMA_F32_16X16X4_F32 (full f32 precision,
// matching the f32 reference exactly in math, only accumulation order moves).
// ---------------------------------------------------------------------------

typedef __attribute__((ext_vector_type(2))) float v2f;
typedef __attribute__((ext_vector_type(8))) float v8f;

#define N_PTS 4096
#define M_OUT 2048
#define FIN   64
#define FOUT  32
#define RSQ   0.25f   // r=0.5 for both ball queries; sqrt(max(d,1e-7))<=0.5 <=> d<=0.25

// ---------------------------------------------------------------------------
// Kernel 1: farthest point sampling. One workgroup (32 waves) per batch.
// xyz cached in LDS (48KB of the 320KB WGP pool); per-thread distances live
// in registers (4 points/thread). Two-level argmax: shfl_xor within wave32,
// then LDS across the 32 waves. Tie-break = smallest index (jnp.argmax).
// Emits fi_xyz directly into the output buffer.
// ---------------------------------------------------------------------------
__global__ __launch_bounds__(1024) void fps_kernel(const float* __restrict__ xyz,
                                                   float* __restrict__ fi_xyz) {
  __shared__ float xs[N_PTS], ys[N_PTS], zs[N_PTS];
  __shared__ float redV[32];
  __shared__ int   redI[32];
  __shared__ int   s_cur;

  const int b    = blockIdx.x;
  const int tid  = threadIdx.x;
  const int lane = tid & 31;
  const int wid  = tid >> 5;

  const float* P = xyz + (size_t)b * N_PTS * 3;
  for (int i = tid; i < N_PTS; i += 1024) {
    xs[i] = P[3 * i + 0];
    ys[i] = P[3 * i + 1];
    zs[i] = P[3 * i + 2];
  }
  __syncthreads();

  // 4 points per thread, register-resident running distances (init 1e5 as ref)
  float d0 = 1e5f, d1 = 1e5f, d2 = 1e5f, d3 = 1e5f;
  const int i0 = tid, i1 = tid + 1024, i2 = tid + 2048, i3 = tid + 3072;
  int cur = 0;  // reference scan starts at index 0

  for (int m = 0; m < M_OUT; ++m) {
    if (tid == 0) {  // record centroid m BEFORE the update (matches lax.scan)
      float* o = fi_xyz + ((size_t)b * M_OUT + m) * 3;
      o[0] = xs[cur]; o[1] = ys[cur]; o[2] = zs[cur];
    }
    const float cx = xs[cur], cy = ys[cur], cz = zs[cur];

    float dx, dy, dz, q;
    dx = xs[i0]-cx; dy = ys[i0]-cy; dz = zs[i0]-cz;
    q = dx*dx + dy*dy + dz*dz; d0 = fminf(d0, sqrtf(fmaxf(q, 1e-7f)));
    dx = xs[i1]-cx; dy = ys[i1]-cy; dz = zs[i1]-cz;
    q = dx*dx + dy*dy + dz*dz; d1 = fminf(d1, sqrtf(fmaxf(q, 1e-7f)));
    dx = xs[i2]-cx; dy = ys[i2]-cy; dz = zs[i2]-cz;
    q = dx*dx + dy*dy + dz*dz; d2 = fminf(d2, sqrtf(fmaxf(q, 1e-7f)));
    dx = xs[i3]-cx; dy = ys[i3]-cy; dz = zs[i3]-cz;
    q = dx*dx + dy*dy + dz*dz; d3 = fminf(d3, sqrtf(fmaxf(q, 1e-7f)));

    // thread-local argmax (strict > keeps the smallest index on ties)
    float bv = d0; int bi = i0;
    if (d1 > bv) { bv = d1; bi = i1; }
    if (d2 > bv) { bv = d2; bi = i2; }
    if (d3 > bv) { bv = d3; bi = i3; }

    // wave32 reduction
    #pragma unroll
    for (int off = 16; off > 0; off >>= 1) {
      float ov = __shfl_xor(bv, off, 32);
      int   oi = __shfl_xor(bi, off, 32);
      if (ov > bv || (ov == bv && oi < bi)) { bv = ov; bi = oi; }
    }
    if (lane == 0) { redV[wid] = bv; redI[wid] = bi; }
    __syncthreads();
    if (wid == 0) {
      bv = redV[lane]; bi = redI[lane];
      #pragma unroll
      for (int off = 16; off > 0; off >>= 1) {
        float ov = __shfl_xor(bv, off, 32);
        int   oi = __shfl_xor(bi, off, 32);
        if (ov > bv || (ov == bv && oi < bi)) { bv = ov; bi = oi; }
      }
      if (lane == 0) s_cur = bi;
    }
    __syncthreads();
    cur = s_cur;
  }
}

// ---------------------------------------------------------------------------
// Ball query (exact reference semantics): first K indices in ascending order
// with dsq <= r^2, padded with the first valid index. Wave-cooperative scan:
// 32 points/iter + __ballot + ctz, early exit once K found.
// ---------------------------------------------------------------------------
__device__ __forceinline__ void ball_scan(const float* __restrict__ P,
                                          float cx, float cy, float cz,
                                          float r2, int K, int* out, int lane) {
  int cnt = 0;
  out[0] = 0;
  for (int base = 0; base < N_PTS && cnt < K; base += 32) {
    const int p = base + lane;
    const float dx = P[3*p+0] - cx;
    const float dy = P[3*p+1] - cy;
    const float dz = P[3*p+2] - cz;
    const bool hit = (dx*dx + dy*dy + dz*dz) <= r2;
    unsigned mask = (unsigned)__ballot(hit);   // wave32: low 32 bits
    while (mask && cnt < K) {
      const int bpos = __builtin_ctz(mask);
      out[cnt++] = base + bpos;
      mask &= mask - 1;
    }
  }
  for (int j = (cnt == 0 ? 1 : cnt); j < K; ++j) out[j] = out[0];
}

// ---------------------------------------------------------------------------
// Kernel 2: per centroid (one wave each): fij = ball_query(xyz, fi_xyz, K=4);
// for each branch t in {1,2,3}: sc = ball_query(xyz, xyz[fij[t]], K=3), then
// Xsum[b,t,m,f] = sum over the 3 (possibly duplicated-pad) neighbors of
// feats[sc, f]. Duplicates DO count (reference sums over k after padding).
// ---------------------------------------------------------------------------
__global__ __launch_bounds__(128) void gather_kernel(const float* __restrict__ xyz,
                                                     const float* __restrict__ rgb,
                                                     const float* __restrict__ fi_xyz,
                                                     float* __restrict__ Xsum) {
  const int w    = blockIdx.x * 4 + (threadIdx.x >> 5);
  const int lane = threadIdx.x & 31;
  const int b    = w >> 11;          // M_OUT == 2048
  const int m    = w & (M_OUT - 1);

  const float* P = xyz + (size_t)b * N_PTS * 3;
  const float* F = rgb + (size_t)b * FIN * N_PTS;   // rgb is (B, FIN, N)
  const float* C = fi_xyz + ((size_t)b * M_OUT + m) * 3;
  const float cx = C[0], cy = C[1], cz = C[2];

  int fij[4];
  ball_scan(P, cx, cy, cz, RSQ, 4, fij, lane);

  #pragma unroll
  for (int t = 1; t <= 3; ++t) {
    const int n = fij[t];
    const float sx = P[3*n+0], sy = P[3*n+1], sz = P[3*n+2];
    int sc[3];
    ball_scan(P, sx, sy, sz, RSQ, 3, sc, lane);

    float* out = Xsum + (((size_t)(b * 3 + (t - 1)) * M_OUT) + m) * FIN;
    #pragma unroll
    for (int h = 0; h < 2; ++h) {
      const int f = lane + 32 * h;   // wave32 covers FIN=64 in two passes
      out[f] = F[(size_t)f * N_PTS + sc[0]]
             + F[(size_t)f * N_PTS + sc[1]]
             + F[(size_t)f * N_PTS + sc[2]];
    }
  }
}

// ---------------------------------------------------------------------------
// Kernel 3: Y = relu(Xsum_t @ para_t) maxed over t, via V_WMMA_F32_16X16X4_F32.
// One wave per 16-centroid row tile: 3 branches x 16 k-steps x 2 n-tiles = 96
// WMMAs. Fragment layouts per ISA 7.12.2:
//   A 16x4 : lanes 0-15 -> M=lane, K={4k,4k+1}; lanes 16-31 -> K={4k+2,4k+3}
//   B 4x16 : VGPR0 rows K=4k(+2hi), VGPR1 rows K=4k+1(+2hi), N = lane%16
//   C/D    : VGPR r -> M = r + 8*(lane>=16), N = lane%16
// Running max initialized to 0 == relu folded into the branch max.
// ---------------------------------------------------------------------------
__global__ __launch_bounds__(128) void wmma_kernel(const float* __restrict__ Xsum,
                                                   const float* __restrict__ p1,
                                                   const float* __restrict__ p2,
                                                   const float* __restrict__ p3,
                                                   float* __restrict__ y) {
  const int w     = blockIdx.x * 4 + (threadIdx.x >> 5);
  const int lane  = threadIdx.x & 31;
  const int b     = w >> 7;            // 128 row tiles per batch
  const int mt    = w & 127;
  const int mbase = mt * 16;
  const int lo    = lane & 15;
  const int hi    = lane >> 4;

  v8f ymax0 = {0.f,0.f,0.f,0.f,0.f,0.f,0.f,0.f};
  v8f ymax1 = {0.f,0.f,0.f,0.f,0.f,0.f,0.f,0.f};

  #pragma unroll
  for (int t = 0; t < 3; ++t) {
    const float* A  = Xsum + (((size_t)(b * 3 + t)) * M_OUT + mbase) * FIN;
    const float* Bp = (t == 0) ? p1 : ((t == 1) ? p2 : p3);   // (FIN, FOUT) row-major

    v8f acc0 = {0.f,0.f,0.f,0.f,0.f,0.f,0.f,0.f};
    v8f acc1 = {0.f,0.f,0.f,0.f,0.f,0.f,0.f,0.f};

    #pragma unroll
    for (int kk = 0; kk < FIN / 4; ++kk) {
      const int kb = kk * 4 + hi * 2;
      v2f a;
      const float* ap = A + lo * FIN + kb;
      a.x = ap[0];
      a.y = ap[1];

      v2f b0, b1;
      b0.x = Bp[(kb + 0) * FOUT + lo];
      b0.y = Bp[(kb + 1) * FOUT + lo];
      b1.x = Bp[(kb + 0) * FOUT + 16 + lo];
      b1.y = Bp[(kb + 1) * FOUT + 16 + lo];

      acc0 = __builtin_amdgcn_wmma_f32_16x16x4_f32(false, a, false, b0,
                                                   (short)0, acc0, false, false);
      acc1 = __builtin_amdgcn_wmma_f32_16x16x4_f32(false, a, false, b1,
                                                   (short)0, acc1, false, false);
    }
    #pragma unroll
    for (int r = 0; r < 8; ++r) {
      ymax0[r] = fmaxf(ymax0[r], acc0[r]);
      ymax1[r] = fmaxf(ymax1[r], acc1[r]);
    }
  }

  // store into y laid out as (B, FOUT, M)
  #pragma unroll
  for (int r = 0; r < 8; ++r) {
    const int mrow = mbase + r + hi * 8;
    y[((size_t)b * FOUT + lo)      * M_OUT + mrow] = ymax0[r];
    y[((size_t)b * FOUT + 16 + lo) * M_OUT + mrow] = ymax1[r];
  }
}

// ---------------------------------------------------------------------------
extern "C" void kernel_launch(void* const* d_in, const int* in_sizes, int n_in,
                              void* d_out, int out_size, void* d_ws, size_t ws_size,
                              hipStream_t stream) {
  const float* rgb = (const float*)d_in[0];   // (B, FIN, N) f32
  const float* xyz = (const float*)d_in[1];   // (B, N, 3)   f32
  const float* p1  = (const float*)d_in[2];   // (FIN, FOUT) f32
  const float* p2  = (const float*)d_in[3];
  const float* p3  = (const float*)d_in[4];

  const int B = in_sizes[0] / (FIN * N_PTS);  // = 2

  float* y      = (float*)d_out;                        // (B, FOUT, M)
  float* fi_xyz = y + (size_t)B * FOUT * M_OUT;         // (B, M, 3)
  float* Xsum   = (float*)d_ws;                         // (B*3, M, FIN) = 3MB

  fps_kernel   <<<B,                1024, 0, stream>>>(xyz, fi_xyz);
  gather_kernel<<<B * M_OUT / 4,     128, 0, stream>>>(xyz, rgb, fi_xyz, Xsum);
  wmma_kernel  <<<B * (M_OUT/16)/4,  128, 0, stream>>>(Xsum, p1, p2, p3, y);
}